// SpectralRemizovLayer_46248207843823
// MI455X (gfx1250) — compile-verified
//
#include <hip/hip_runtime.h>
#include <math.h>

#define DEV __device__ __forceinline__

typedef __attribute__((ext_vector_type(2))) float v2f;
typedef __attribute__((ext_vector_type(8))) float v8f;
typedef __attribute__((address_space(3))) float lds_f;

// Problem sizes
constexpr int B    = 16;
constexpr int C    = 128;
constexpr int H    = 128;
constexpr int W    = 128;
constexpr int M1   = 16;   // modes (H dim)
constexpr int M2   = 16;   // modes (W dim)
constexpr int RANK = 8;
constexpr int MM   = M1 * M2;        // 256
constexpr int NIMG = B * C;          // 2048

// Workspace layout (float offsets)
constexpr size_t OFF_FC  = 0;                       // [128][16] fwd W cos /sqrt(128)
constexpr size_t OFF_FS  = OFF_FC  + 2048;          // [128][16] fwd W  -sin/sqrt(128)
constexpr size_t OFF_HC  = OFF_FS  + 2048;          // [16][128] fwd H cos /sqrt(128)
constexpr size_t OFF_HS  = OFF_HC  + 2048;          // [16][128] fwd H sin /sqrt(128)
constexpr size_t OFF_EC  = OFF_HS  + 2048;          // [16][128] inv H cos
constexpr size_t OFF_ES  = OFF_EC  + 2048;          // [16][128] inv H sin
constexpr size_t OFF_GC  = OFF_ES  + 2048;          // [16][128] inv W c_k*cos/128
constexpr size_t OFF_GS  = OFF_GC  + 2048;          // [16][128] inv W c_k*sin/128
constexpr size_t OFF_YR  = OFF_GS  + 2048;          // x_top real  [2048][256]
constexpr size_t OFF_YI  = OFF_YR  + (size_t)NIMG*MM;
constexpr size_t OFF_Y2R = OFF_YI  + (size_t)NIMG*MM; // out_top real
constexpr size_t OFF_Y2I = OFF_Y2R + (size_t)NIMG*MM;
constexpr size_t OFF_ZR  = OFF_Y2I + (size_t)NIMG*MM; // z real [16][8]
constexpr size_t OFF_ZI  = OFF_ZR  + 128;
constexpr size_t OFF_TR  = OFF_ZI  + 128;             // T real [2048][16][128]
constexpr size_t OFF_TI  = OFF_TR  + (size_t)NIMG*M2*H;

// ---------- WMMA helpers (V_WMMA_F32_16X16X4_F32; layouts per ISA 7.12.2) ----------
// A 16x4 f32: lane l: M = l&15, half = l>>4; vgpr0 = elem(M, 2*half), vgpr1 = elem(M, 2*half+1)
// B 4x16 f32: lane l: N = l&15, half = l>>4; vgpr0 = elem(2*half, N), vgpr1 = elem(2*half+1, N)
// C/D 16x16:  vgpr v: (M = v + 8*half, N = l&15)
DEV v2f frag(const float* p, int ms, int ks, int mn, int k0, int half) {
  int k = k0 + 2 * half;
  v2f r;
  r.x = p[(size_t)mn * ms + (size_t)k * ks];
  r.y = p[(size_t)mn * ms + (size_t)(k + 1) * ks];
  return r;
}
DEV v2f negf(v2f a) { v2f r; r.x = -a.x; r.y = -a.y; return r; }
DEV v8f wmma4(v2f a, v2f b, v8f c) {
  return __builtin_amdgcn_wmma_f32_16x16x4_f32(false, a, false, b, (short)0, c,
                                               false, false);
}

// ---------- async memory -> LDS helpers (CDNA5 GLOBAL_LOAD_ASYNC_TO_LDS_*) ----
DEV unsigned lds_off(const void* p) {
  return (unsigned)(unsigned long long)(lds_f*)p;  // addrspace(3) byte offset
}
DEV void async_ld_b128(unsigned lds_byte, const void* gaddr) {
  asm volatile("global_load_async_to_lds_b128 %0, %1, off"
               :
               : "v"(lds_byte), "v"((unsigned long long)(size_t)gaddr)
               : "memory");
}
DEV void async_ld_b32(unsigned lds_byte, const void* gaddr) {
  asm volatile("global_load_async_to_lds_b32 %0, %1, off"
               :
               : "v"(lds_byte), "v"((unsigned long long)(size_t)gaddr)
               : "memory");
}
DEV void wait_async0() { asm volatile("s_wait_asynccnt 0x0" ::: "memory"); }

// ---------- kernel 0: trig tables ----------
__global__ __launch_bounds__(256) void k_init(float* ws) {
  int idx = blockIdx.x * 256 + threadIdx.x;
  if (idx >= 16 * 128) return;
  int a = idx >> 7;   // frequency 0..15
  int p = idx & 127;  // position  0..127
  int ph = (a * p) & 127;
  float theta = (2.0f * 3.14159265358979323846f / 128.0f) * (float)ph;
  float s, c;
  __sincosf(theta, &s, &c);
  const float inv = 0.08838834764831845f;  // 1/sqrt(128)
  ws[OFF_FC + p * 16 + a] = c * inv;
  ws[OFF_FS + p * 16 + a] = -s * inv;
  ws[OFF_HC + a * 128 + p] = c * inv;
  ws[OFF_HS + a * 128 + p] = s * inv;
  ws[OFF_EC + a * 128 + p] = c;
  ws[OFF_ES + a * 128 + p] = s;
  float ck = (a == 0) ? 1.0f : 2.0f;
  ws[OFF_GC + a * 128 + p] = ck * c * (1.0f / 128.0f);
  ws[OFF_GS + a * 128 + p] = ck * s * (1.0f / 128.0f);
}

// ---------- kernel 1: forward partial DFT: x -> x_top (complex 16x16) ----------
__global__ __launch_bounds__(256) void k_fwd(const float* __restrict__ x,
                                             float* __restrict__ ws) {
  __shared__ float sTr[H * M2];
  __shared__ float sTi[H * M2];
  const float* Fc = ws + OFF_FC;
  const float* Fs = ws + OFF_FS;
  const float* Hc = ws + OFF_HC;
  const float* Hs = ws + OFF_HS;
  float* Yr = ws + OFF_YR;
  float* Yi = ws + OFF_YI;

  int img = blockIdx.x;  // b*C + c
  int wid = threadIdx.x >> 5, l = threadIdx.x & 31;
  int half = l >> 4, mn = l & 15;
  const float* ximg = x + (size_t)img * H * W;

  // stage 1: T[h, kw] over 16-row tile per wave
  int r0 = wid * 16;
  v8f aTr = {}, aTi = {};
  for (int k0 = 0; k0 < W; k0 += 4) {
    v2f xa = frag(ximg + (size_t)r0 * W, W, 1, mn, k0, half);  // A: elem(h,w)
    v2f bc = frag(Fc, 1, 16, mn, k0, half);                    // B: elem(w,kw)
    v2f bs = frag(Fs, 1, 16, mn, k0, half);
    aTr = wmma4(xa, bc, aTr);
    aTi = wmma4(xa, bs, aTi);
  }
  for (int v = 0; v < 8; ++v) {
    int h = r0 + v + 8 * half;
    sTr[h * 16 + mn] = aTr[v];
    sTi[h * 16 + mn] = aTi[v];
  }
  __syncthreads();

  // stage 2: Y[kh,kw] = sum_h e^{-i th}(Tr + i Ti); wave0 -> Yr, wave1 -> Yi
  if (wid < 2) {
    v8f acc = {};
    for (int k0 = 0; k0 < H; k0 += 4) {
      v2f hc = frag(Hc, 128, 1, mn, k0, half);  // A: elem(kh,h)
      v2f hs = frag(Hs, 128, 1, mn, k0, half);
      v2f tr = frag(sTr, 1, 16, mn, k0, half);  // B: elem(h,kw)
      v2f ti = frag(sTi, 1, 16, mn, k0, half);
      if (wid == 0) {
        acc = wmma4(hc, tr, acc);
        acc = wmma4(hs, ti, acc);
      } else {
        acc = wmma4(hc, ti, acc);
        acc = wmma4(negf(hs), tr, acc);
      }
    }
    float* dst = (wid == 0 ? Yr : Yi) + (size_t)img * MM;
    for (int v = 0; v < 8; ++v) dst[(v + 8 * half) * 16 + mn] = acc[v];
  }
}

// ---------- kernel 2a: z[b,r] = sum_{i,m,n} x_top * conj(V) ----------
__global__ __launch_bounds__(256) void k_z(const float* __restrict__ vr,
                                           const float* __restrict__ vi,
                                           float* __restrict__ ws) {
  int b = blockIdx.x >> 3, r = blockIdx.x & 7;
  const float* Yr = ws + OFF_YR;
  const float* Yi = ws + OFF_YI;
  float re = 0.f, im = 0.f;
  for (int t = threadIdx.x; t < C * MM; t += 256) {
    int i = t >> 8, mode = t & 255;
    size_t xo = ((size_t)(b * C + i)) * MM + mode;
    size_t vo = ((size_t)(i * RANK + r)) * MM + mode;
    float xr = Yr[xo], xi = Yi[xo];
    float vre = vr[vo], vim = vi[vo];
    re += xr * vre + xi * vim;
    im += xi * vre - xr * vim;
  }
  __shared__ float sre[256], sim[256];
  int t = threadIdx.x;
  sre[t] = re;
  sim[t] = im;
  __syncthreads();
  for (int s = 128; s > 0; s >>= 1) {
    if (t < s) {
      sre[t] += sre[t + s];
      sim[t] += sim[t + s];
    }
    __syncthreads();
  }
  if (t == 0) {
    ws[OFF_ZR + blockIdx.x] = sre[0];
    ws[OFF_ZI + blockIdx.x] = sim[0];
  }
}

// ---------- kernel 2b: per-mode complex GEMM + low-rank correction ----------
__global__ __launch_bounds__(256) void k_mode(const float* __restrict__ wr,
                                              const float* __restrict__ wim,
                                              const float* __restrict__ ur,
                                              const float* __restrict__ ui,
                                              float* __restrict__ ws) {
  __shared__ __align__(16) float sAr[B * C];  // [b][i]
  __shared__ __align__(16) float sAi[B * C];
  int mode = blockIdx.x;
  int wid = threadIdx.x >> 5, l = threadIdx.x & 31;
  int half = l >> 4, mn = l & 15;
  const float* Yr = ws + OFF_YR;
  const float* Yi = ws + OFF_YI;
  // async gather (stride-1KB elements): memory -> LDS without VGPR round-trip
  {
    unsigned ar0 = lds_off(sAr), ai0 = lds_off(sAi);
    for (int t = threadIdx.x; t < B * C; t += 256) {
      size_t off = (size_t)t * MM + mode;  // t = bb*C + i
      async_ld_b32(ar0 + (unsigned)t * 4, Yr + off);
      async_ld_b32(ai0 + (unsigned)t * 4, Yi + off);
    }
    wait_async0();
  }
  __syncthreads();

  int o0 = wid * 16;
  v8f accR = {}, accI = {};
  for (int k0 = 0; k0 < C; k0 += 4) {
    v2f ar = frag(sAr, 128, 1, mn, k0, half);  // A: elem(b,i)
    v2f ai = frag(sAi, 128, 1, mn, k0, half);
    // B: elem(i,o) = w[((o0+n)*C + i)*256 + mode]
    v2f br = frag(wr + mode, C * MM, MM, o0 + mn, k0, half);
    v2f bi = frag(wim + mode, C * MM, MM, o0 + mn, k0, half);
    accR = wmma4(ar, br, accR);
    accR = wmma4(negf(ai), bi, accR);
    accI = wmma4(ar, bi, accI);
    accI = wmma4(ai, br, accI);
  }

  const float* zr = ws + OFF_ZR;
  const float* zi = ws + OFF_ZI;
  float* Y2r = ws + OFF_Y2R;
  float* Y2i = ws + OFF_Y2I;
  int o = o0 + mn;
  for (int v = 0; v < 8; ++v) {
    int bb = v + 8 * half;
    float addr_ = 0.f, addi_ = 0.f;
    for (int r = 0; r < RANK; ++r) {
      size_t uo = ((size_t)(o * RANK + r)) * MM + mode;
      float u_r = ur[uo], u_i = ui[uo];
      float z_r = zr[bb * RANK + r], z_i = zi[bb * RANK + r];
      addr_ += u_r * z_r - u_i * z_i;
      addi_ += u_r * z_i + u_i * z_r;
    }
    size_t oo = ((size_t)(bb * C + o)) * MM + mode;
    Y2r[oo] = accR[v] + addr_;
    Y2i[oo] = accI[v] + addi_;
  }
}

// ---------- kernel 3: inverse DFT stage 1: T[kw,h] per (b,o) ----------
__global__ __launch_bounds__(256) void k_ist1(float* __restrict__ ws) {
  int img = blockIdx.x;  // b*C + o
  int wid = threadIdx.x >> 5, l = threadIdx.x & 31;
  int half = l >> 4, mn = l & 15;
  const float* yr = ws + OFF_Y2R + (size_t)img * MM;
  const float* yi = ws + OFF_Y2I + (size_t)img * MM;
  const float* Ec = ws + OFF_EC;
  const float* Es = ws + OFF_ES;
  int h0 = wid * 16;
  v8f aTr = {}, aTi = {};
  for (int k0 = 0; k0 < M1; k0 += 4) {
    v2f fyr = frag(yr, 1, 16, mn, k0, half);       // A: elem(kw,kh) = y[kh*16+kw]
    v2f fyi = frag(yi, 1, 16, mn, k0, half);
    v2f ec = frag(Ec + h0, 1, 128, mn, k0, half);  // B: elem(kh,h)
    v2f es = frag(Es + h0, 1, 128, mn, k0, half);
    aTr = wmma4(fyr, ec, aTr);
    aTr = wmma4(negf(fyi), es, aTr);
    aTi = wmma4(fyr, es, aTi);
    aTi = wmma4(fyi, ec, aTi);
  }
  float* tr = ws + OFF_TR + (size_t)img * (M2 * H) + h0 + mn;
  float* ti = ws + OFF_TI + (size_t)img * (M2 * H) + h0 + mn;
  for (int v = 0; v < 8; ++v) {
    tr[(v + 8 * half) * 128] = aTr[v];
    ti[(v + 8 * half) * 128] = aTi[v];
  }
}

// ---------- kernel 4: fused skip GEMM + inverse stage 2 + activation ----------
__global__ __launch_bounds__(256) void k_out(const float* __restrict__ x,
                                             const float* __restrict__ sw,
                                             const float* __restrict__ ws,
                                             float* __restrict__ out) {
  __shared__ __align__(16) float xs[C * 16];  // [c][w]
  int wt = blockIdx.x & 7;
  int bh = blockIdx.x >> 3;
  int h = bh & 127;
  int b = bh >> 7;
  int w0 = wt * 16;
  int wid = threadIdx.x >> 5, l = threadIdx.x & 31;
  int half = l >> 4, mn = l & 15;
  int o0 = wid * 16;

  __builtin_prefetch(sw + (size_t)o0 * C, 0, 3);  // skip_w rows for this wave

  // async stage of the 128x16 x tile: 512 16-byte chunks, no VGPR round-trip
  {
    unsigned x0 = lds_off(xs);
    const float* base = x + (((size_t)(b * C)) * H + h) * W + w0;
    for (int e = threadIdx.x; e < 512; e += 256) {
      int c = e >> 2, wq = (e & 3) * 4;
      async_ld_b128(x0 + (unsigned)e * 16, base + (size_t)c * H * W + wq);
    }
    wait_async0();
  }
  __syncthreads();

  v8f acc = {};
  // skip GEMM: out[o,w] += sum_c skip_w[o,c] * x[c,w]
  for (int k0 = 0; k0 < C; k0 += 4) {
    v2f a = frag(sw + (size_t)o0 * C, C, 1, mn, k0, half);  // A: elem(o,c)
    v2f bx = frag(xs, 1, 16, mn, k0, half);                 // B: elem(c,w)
    acc = wmma4(a, bx, acc);
  }
  // inverse stage 2: out[o,w] += sum_kw c_kw*(Tr*cos - Ti*sin)
  const float* Gc = ws + OFF_GC;
  const float* Gs = ws + OFF_GS;
  const float* trp = ws + OFF_TR + ((size_t)(b * C + o0)) * (M2 * H) + h;
  const float* tip = ws + OFF_TI + ((size_t)(b * C + o0)) * (M2 * H) + h;
  for (int k0 = 0; k0 < M2; k0 += 4) {
    v2f tr = frag(trp, M2 * H, 128, mn, k0, half);  // A: elem(o,kw)
    v2f ti = frag(tip, M2 * H, 128, mn, k0, half);
    v2f gc = frag(Gc + w0, 1, 128, mn, k0, half);   // B: elem(kw,w)
    v2f gs = frag(Gs + w0, 1, 128, mn, k0, half);
    acc = wmma4(tr, gc, acc);
    acc = wmma4(negf(ti), gs, acc);
  }
  // activation s*exp(-s^2) + store
  for (int v = 0; v < 8; ++v) {
    int o = o0 + v + 8 * half;
    float s = acc[v];
    out[(((size_t)(b * C + o)) * H + h) * W + w0 + mn] = s * __expf(-s * s);
  }
}

extern "C" void kernel_launch(void* const* d_in, const int* in_sizes, int n_in,
                              void* d_out, int out_size, void* d_ws,
                              size_t ws_size, hipStream_t stream) {
  const float* x = (const float*)d_in[0];
  const float* w_real = (const float*)d_in[1];
  const float* w_imag = (const float*)d_in[2];
  const float* u_real = (const float*)d_in[3];
  const float* u_imag = (const float*)d_in[4];
  const float* v_real = (const float*)d_in[5];
  const float* v_imag = (const float*)d_in[6];
  const float* skip_w = (const float*)d_in[7];
  float* out = (float*)d_out;
  float* ws = (float*)d_ws;

  k_init<<<8, 256, 0, stream>>>(ws);
  k_fwd<<<NIMG, 256, 0, stream>>>(x, ws);
  k_z<<<B * RANK, 256, 0, stream>>>(v_real, v_imag, ws);
  k_mode<<<MM, 256, 0, stream>>>(w_real, w_imag, u_real, u_imag, ws);
  k_ist1<<<NIMG, 256, 0, stream>>>(ws);
  k_out<<<B * H * (W / 16), 256, 0, stream>>>(x, skip_w, ws, out);
}